// NeuromorphicSpikingNetwork_12463995093891
// MI455X (gfx1250) — compile-verified
//
#include <hip/hip_runtime.h>
#include <cmath>

// Problem constants (match reference)
#define BB   16
#define DD   2048
#define TT   100
#define NN0  4096
#define NN1  4096
#define NN2  2048
#define NCLS 10

typedef __attribute__((ext_vector_type(2))) float v2f;
typedef __attribute__((ext_vector_type(8))) float v8f;

// D = A(16x4) * B(4x16) + C(16x16), all fp32 (exact) -> v_wmma_f32_16x16x4_f32
__device__ __forceinline__ v8f wmma4(v2f a, v2f b, v8f c) {
  return __builtin_amdgcn_wmma_f32_16x16x4_f32(false, a, false, b, (short)0, c,
                                               false, false);
}

// ---------------------------------------------------------------------------
// Shared forward-GEMM tile: one block (8 waves) computes a 16(batch) x 16(col)
// tile of A(16xK) @ W(KxN). Each wave covers K/8, LDS-reduce, wave 0 gets the
// 8 per-lane sums (C-layout: m = r + 8*(lane>=16), n = lane&15).
// ---------------------------------------------------------------------------
template <int K, int N>
__device__ __forceinline__ bool tile_gemm(const float* __restrict__ A,
                                          const float* __restrict__ W,
                                          float* __restrict__ red,
                                          float sums[8]) {
  const int lane = threadIdx.x & 31;
  const int wave = threadIdx.x >> 5;
  const int half = lane >> 4;
  const int l16  = lane & 15;
  const int kb   = half * 2;
  const int j0   = blockIdx.x * 16;

  v8f acc = {};
  const int kchunk = K / 8;
  const int kstart = wave * kchunk;
  const float* arow = A + l16 * K;  // A row = batch index l16
#pragma unroll 4
  for (int k0 = kstart; k0 < kstart + kchunk; k0 += 4) {
    v2f a, b;
    a.x = arow[k0 + kb];
    a.y = arow[k0 + kb + 1];
    const float* bp = W + (size_t)(k0 + kb) * N + j0 + l16;
    b.x = bp[0];
    b.y = bp[N];
    acc = wmma4(a, b, acc);
  }
#pragma unroll
  for (int r = 0; r < 8; ++r) red[wave * 256 + r * 32 + lane] = acc[r];
  __syncthreads();
  if (wave == 0) {
#pragma unroll
    for (int r = 0; r < 8; ++r) {
      float s = 0.0f;
#pragma unroll
      for (int w = 0; w < 8; ++w) s += red[w * 256 + r * 32 + lane];
      sums[r] = s;
    }
    return true;
  }
  return false;
}

// ---------------------------------------------------------------------------
// Encoder: rates = sigmoid(x @ W_enc + b_enc)   (16x2048)@(2048x4096)
// ---------------------------------------------------------------------------
__global__ __launch_bounds__(256) void encoder_kernel(
    const float* __restrict__ x, const float* __restrict__ W_enc,
    const float* __restrict__ b_enc, float* __restrict__ rates) {
  __shared__ float red[2048];
  float sums[8];
  if (tile_gemm<DD, NN0>(x, W_enc, red, sums)) {
    const int lane = threadIdx.x & 31;
    const int half = lane >> 4, l16 = lane & 15;
    const int j = blockIdx.x * 16 + l16;
    const float bj = b_enc[j];
#pragma unroll
    for (int r = 0; r < 8; ++r) {
      const int b = r + 8 * half;
      const float z = sums[r] + bj;
      rates[b * NN0 + j] = 1.0f / (1.0f + __expf(-z));
    }
  }
}

// ---------------------------------------------------------------------------
// LIF layer 0: bernoulli spike encode + LIF + trace tr0 (= p0)
// ---------------------------------------------------------------------------
__global__ __launch_bounds__(256) void lif0_kernel(
    const float* __restrict__ u, const float* __restrict__ rates,
    float* __restrict__ V0, float* __restrict__ rf0, float* __restrict__ s0,
    float* __restrict__ tr0, int t) {
  const int i = blockIdx.x * blockDim.x + threadIdx.x;  // over B*N0
  const int b = i >> 12;          // / NN0
  const int n = i & (NN0 - 1);
  const float sp_in = (u[(size_t)b * TT * NN0 + (size_t)t * NN0 + n] < rates[i])
                          ? 1.0f : 0.0f;
  float V = 0.9f * V0[i] + sp_in;
  const float rr = rf0[i];
  const bool spk = (V > 1.0f) && (rr <= 0.0f);
  const float s = spk ? 1.0f : 0.0f;
  V0[i]  = spk ? 0.0f : V;
  rf0[i] = spk ? 2.0f : fmaxf(rr - 1.0f, 0.0f);
  s0[i]  = s;
  tr0[i] = 0.95f * tr0[i] + s;
}

// ---------------------------------------------------------------------------
// fc1: c1 = s0 @ Wa, fused LIF1 + trace tr1 (= q0 = p1)
// ---------------------------------------------------------------------------
__global__ __launch_bounds__(256) void fc1_lif_kernel(
    const float* __restrict__ s0, const float* __restrict__ Wa,
    float* __restrict__ V1, float* __restrict__ rf1, float* __restrict__ s1,
    float* __restrict__ tr1) {
  __shared__ float red[2048];
  float sums[8];
  if (tile_gemm<NN0, NN1>(s0, Wa, red, sums)) {
    const int lane = threadIdx.x & 31;
    const int half = lane >> 4, l16 = lane & 15;
    const int j = blockIdx.x * 16 + l16;
#pragma unroll
    for (int r = 0; r < 8; ++r) {
      const int b   = r + 8 * half;
      const int idx = b * NN1 + j;
      float V = 0.9f * V1[idx] + sums[r];
      const float rr = rf1[idx];
      const bool spk = (V > 1.0f) && (rr <= 0.0f);
      const float s = spk ? 1.0f : 0.0f;
      V1[idx]  = spk ? 0.0f : V;
      rf1[idx] = spk ? 2.0f : fmaxf(rr - 1.0f, 0.0f);
      s1[idx]  = s;
      tr1[idx] = 0.95f * tr1[idx] + s;
    }
  }
}

// ---------------------------------------------------------------------------
// fc2: c2 = s1 @ Wb, fused LIF2 + trace tr2 (= q1) + spike counts
// ---------------------------------------------------------------------------
__global__ __launch_bounds__(256) void fc2_lif_kernel(
    const float* __restrict__ s1, const float* __restrict__ Wb,
    float* __restrict__ V2, float* __restrict__ rf2, float* __restrict__ s2,
    float* __restrict__ tr2, float* __restrict__ counts) {
  __shared__ float red[2048];
  float sums[8];
  if (tile_gemm<NN1, NN2>(s1, Wb, red, sums)) {
    const int lane = threadIdx.x & 31;
    const int half = lane >> 4, l16 = lane & 15;
    const int j = blockIdx.x * 16 + l16;
#pragma unroll
    for (int r = 0; r < 8; ++r) {
      const int b   = r + 8 * half;
      const int idx = b * NN2 + j;
      float V = 0.9f * V2[idx] + sums[r];
      const float rr = rf2[idx];
      const bool spk = (V > 1.0f) && (rr <= 0.0f);
      const float s = spk ? 1.0f : 0.0f;
      V2[idx]  = spk ? 0.0f : V;
      rf2[idx] = spk ? 2.0f : fmaxf(rr - 1.0f, 0.0f);
      s2[idx]  = s;
      tr2[idx] = 0.95f * tr2[idx] + s;
      counts[idx] += s;
    }
  }
}

// ---------------------------------------------------------------------------
// STDP: W(tile) = clamp(W + LR*(trPre^T @ sPost - sPre^T @ trPost), -1, 1)
// One wave per 16x16 weight tile; K = B = 16 -> 4+4 fp32 WMMAs per tile.
// ---------------------------------------------------------------------------
__global__ __launch_bounds__(256) void stdp_kernel(
    float* __restrict__ W, const float* __restrict__ trPre,
    const float* __restrict__ sPre, const float* __restrict__ sPost,
    const float* __restrict__ trPost, int Npre, int Npost) {
  const int lane = threadIdx.x & 31;
  const int wave = threadIdx.x >> 5;
  const int tile = blockIdx.x * 8 + wave;
  const int ntj  = Npost >> 4;
  const int it = tile / ntj, jt = tile % ntj;
  const int i0 = it * 16, j0 = jt * 16;
  const int half = lane >> 4, l16 = lane & 15, kb = half * 2;

  v8f accP = {}, accM = {};
#pragma unroll
  for (int b0 = 0; b0 < BB; b0 += 4) {
    v2f a, b;
    a.x = trPre[(b0 + kb) * Npre + i0 + l16];
    a.y = trPre[(b0 + kb + 1) * Npre + i0 + l16];
    b.x = sPost[(b0 + kb) * Npost + j0 + l16];
    b.y = sPost[(b0 + kb + 1) * Npost + j0 + l16];
    accP = wmma4(a, b, accP);
    a.x = sPre[(b0 + kb) * Npre + i0 + l16];
    a.y = sPre[(b0 + kb + 1) * Npre + i0 + l16];
    b.x = trPost[(b0 + kb) * Npost + j0 + l16];
    b.y = trPost[(b0 + kb + 1) * Npost + j0 + l16];
    accM = wmma4(a, b, accM);
  }
#pragma unroll
  for (int r = 0; r < 8; ++r) {
    const int m = r + 8 * half;
    const size_t idx = (size_t)(i0 + m) * Npost + j0 + l16;
    const float w = W[idx] + 0.01f * (accP[r] - accM[r]);
    W[idx] = fminf(fmaxf(w, -1.0f), 1.0f);
  }
}

// ---------------------------------------------------------------------------
// Decoder: out = counts @ W_dec + b_dec   (16x2048)@(2048x10)
// ---------------------------------------------------------------------------
__global__ void decode_kernel(const float* __restrict__ counts,
                              const float* __restrict__ Wd,
                              const float* __restrict__ bd,
                              float* __restrict__ out) {
  const int i = threadIdx.x;
  if (i >= BB * NCLS) return;
  const int b = i / NCLS, c = i % NCLS;
  float sum = bd[c];
  for (int k = 0; k < NN2; ++k) sum += counts[b * NN2 + k] * Wd[k * NCLS + c];
  out[i] = sum;
}

__global__ void zero_kernel(float* __restrict__ p, int n) {
  const int i = blockIdx.x * blockDim.x + threadIdx.x;
  if (i < n) p[i] = 0.0f;
}

// ---------------------------------------------------------------------------
extern "C" void kernel_launch(void* const* d_in, const int* in_sizes, int n_in,
                              void* d_out, int out_size, void* d_ws,
                              size_t ws_size, hipStream_t stream) {
  const float* x     = (const float*)d_in[0];
  const float* u     = (const float*)d_in[1];
  const float* W_enc = (const float*)d_in[2];
  const float* b_enc = (const float*)d_in[3];
  const float* W0    = (const float*)d_in[4];
  const float* W1    = (const float*)d_in[5];
  const float* W_dec = (const float*)d_in[6];
  const float* b_dec = (const float*)d_in[7];
  float* out = (float*)d_out;

  // Workspace carve-up (fp32 everywhere). Wa+Wb = 96MB (L2-resident on MI455X).
  float* p      = (float*)d_ws;
  float* Wa     = p; p += (size_t)NN0 * NN1;
  float* Wb     = p; p += (size_t)NN1 * NN2;
  float* rates  = p; p += BB * NN0;
  float* zbase  = p;
  float* V0     = p; p += BB * NN0;
  float* rf0    = p; p += BB * NN0;
  float* tr0    = p; p += BB * NN0;
  float* s0     = p; p += BB * NN0;
  float* V1     = p; p += BB * NN1;
  float* rf1    = p; p += BB * NN1;
  float* tr1    = p; p += BB * NN1;
  float* s1     = p; p += BB * NN1;
  float* V2     = p; p += BB * NN2;
  float* rf2    = p; p += BB * NN2;
  float* tr2    = p; p += BB * NN2;
  float* s2     = p; p += BB * NN2;
  float* counts = p; p += BB * NN2;
  const int nzero = (int)(p - zbase);

  // Fresh mutable weights each call (inputs never modified; deterministic).
  hipMemcpyAsync(Wa, W0, sizeof(float) * (size_t)NN0 * NN1,
                 hipMemcpyDeviceToDevice, stream);
  hipMemcpyAsync(Wb, W1, sizeof(float) * (size_t)NN1 * NN2,
                 hipMemcpyDeviceToDevice, stream);
  zero_kernel<<<(nzero + 255) / 256, 256, 0, stream>>>(zbase, nzero);

  encoder_kernel<<<NN0 / 16, 256, 0, stream>>>(x, W_enc, b_enc, rates);

  for (int t = 0; t < TT; ++t) {
    lif0_kernel<<<(BB * NN0) / 256, 256, 0, stream>>>(u, rates, V0, rf0, s0,
                                                      tr0, t);
    fc1_lif_kernel<<<NN1 / 16, 256, 0, stream>>>(s0, Wa, V1, rf1, s1, tr1);
    fc2_lif_kernel<<<NN2 / 16, 256, 0, stream>>>(s1, Wb, V2, rf2, s2, tr2,
                                                 counts);
    stdp_kernel<<<(NN0 / 16) * (NN1 / 16) / 8, 256, 0, stream>>>(
        Wa, tr0, s0, s1, tr1, NN0, NN1);
    stdp_kernel<<<(NN1 / 16) * (NN2 / 16) / 8, 256, 0, stream>>>(
        Wb, tr1, s1, s2, tr2, NN1, NN2);
  }

  decode_kernel<<<1, 192, 0, stream>>>(counts, W_dec, b_dec, out);
}